// SemanticAssignmentModule_85005992722930
// MI455X (gfx1250) — compile-verified
//
#include <hip/hip_runtime.h>
#include <hip/hip_bf16.h>
#include <cmath>

typedef __attribute__((ext_vector_type(2))) float v2f;
typedef __attribute__((ext_vector_type(8))) float v8f;

// ---------------------------------------------------------------------------
// Problem constants
// ---------------------------------------------------------------------------
#define NPOS   4096      // N = (H/2)*(W/2)
#define DDIM   128
#define NCLS   1000

// output offsets (floats) in return order: g_tl, C, f_s1, f_s2, f_s3
#define O_GTL  0
#define O_C    1000
#define O_FS1  (O_C   + NPOS*NPOS)        // 8*8*512
#define O_FS2  (O_FS1 + 8*8*512)          // 16*16*256
#define O_FS3  (O_FS2 + 16*16*256)        // 32*32*128

// workspace offsets (floats)
#define W_TN    0
#define W_RN    (W_TN  + NPOS*DDIM)
#define W_F1    (W_RN  + NPOS*DDIM)       // conv1 out 128*128*64
#define W_F     (W_F1  + 128*128*64)      // conv2 out 64*64*128
#define W_FA    (W_F   + 64*64*128)       // attention out 4096*128
#define W_RMAX  (W_FA  + NPOS*DDIM)
#define W_RINV  (W_RMAX + NPOS)
#define W_PM    (W_RINV + NPOS)           // 16*128 partial col-max
#define W_G     (W_PM  + 16*128)          // 512

// LDS byte offset (what async-to-LDS VDST expects) from a generic shared ptr
__device__ __forceinline__ unsigned lds_off(const void* p) {
  return (unsigned)(size_t)(__attribute__((address_space(3))) const void*)p;
}

// ---------------------------------------------------------------------------
// 1) Row L2-normalize: one wave (32 lanes) per 128-float row
// ---------------------------------------------------------------------------
__global__ __launch_bounds__(256) void rownorm_kernel(const float* __restrict__ x,
                                                      float* __restrict__ y, int nrows) {
  int wave = (blockIdx.x * blockDim.x + threadIdx.x) >> 5;
  int lane = threadIdx.x & 31;
  if (wave >= nrows) return;
  const float4* src = reinterpret_cast<const float4*>(x + (size_t)wave * DDIM);
  float4 v = src[lane];
  float ss = v.x * v.x + v.y * v.y + v.z * v.z + v.w * v.w;
#pragma unroll
  for (int off = 16; off > 0; off >>= 1) ss += __shfl_xor(ss, off, 32);
  float inv = 1.0f / (sqrtf(ss) + 1e-8f);
  v.x *= inv; v.y *= inv; v.z *= inv; v.w *= inv;
  reinterpret_cast<float4*>(y + (size_t)wave * DDIM)[lane] = v;
}

// ---------------------------------------------------------------------------
// 2) C = A @ B^T with A,B row-major [4096 x 128] (both L2-resident).
//    WMMA f32 16x16x4; 8 waves cover a 128x128 C tile, each wave 32x64.
// ---------------------------------------------------------------------------
__global__ __launch_bounds__(256) void gemm_nt_wmma_kernel(const float* __restrict__ A,
                                                           const float* __restrict__ B,
                                                           float* __restrict__ C, int ldc) {
  const int tid  = threadIdx.x;
  const int w    = tid >> 5;
  const int lane = tid & 31;
  const int lo   = lane & 15;
  const int hi   = lane >> 4;
  const int wm   = (w & 3) * 32;
  const int wn   = (w >> 2) * 64;
  const int bm   = blockIdx.y * 128;
  const int bn   = blockIdx.x * 128;

  v8f zero;
#pragma unroll
  for (int e = 0; e < 8; ++e) zero[e] = 0.0f;
  v8f acc[2][4];
#pragma unroll
  for (int mi = 0; mi < 2; ++mi)
#pragma unroll
    for (int ni = 0; ni < 4; ++ni) acc[mi][ni] = zero;

  const float* Arow0 = A + (size_t)(bm + wm + lo) * DDIM;
  const float* Brow0 = B + (size_t)(bn + wn + lo) * DDIM;

#pragma unroll 4
  for (int kc = 0; kc < DDIM; kc += 4) {
    const int kk = kc + 2 * hi;
    v2f a[2], b[4];
#pragma unroll
    for (int mi = 0; mi < 2; ++mi)
      a[mi] = *reinterpret_cast<const v2f*>(Arow0 + (size_t)(16 * mi) * DDIM + kk);
#pragma unroll
    for (int ni = 0; ni < 4; ++ni)
      b[ni] = *reinterpret_cast<const v2f*>(Brow0 + (size_t)(16 * ni) * DDIM + kk);
#pragma unroll
    for (int mi = 0; mi < 2; ++mi)
#pragma unroll
      for (int ni = 0; ni < 4; ++ni)
        acc[mi][ni] = __builtin_amdgcn_wmma_f32_16x16x4_f32(
            false, a[mi], false, b[ni], (short)0, acc[mi][ni], false, false);
  }

#pragma unroll
  for (int mi = 0; mi < 2; ++mi)
#pragma unroll
    for (int ni = 0; ni < 4; ++ni) {
      const int col = bn + wn + 16 * ni + lo;
#pragma unroll
      for (int r = 0; r < 8; ++r) {
        const int row = bm + wm + 16 * mi + r + 8 * hi;
        C[(size_t)row * ldc + col] = acc[mi][ni][r];
      }
    }
}

// ---------------------------------------------------------------------------
// 3) Per-row softmax stats over C (L2-resident): rowmax, 1/sum(exp(c-max)).
// ---------------------------------------------------------------------------
__global__ __launch_bounds__(256) void rowstats_kernel(const float* __restrict__ C,
                                                       float* __restrict__ rmax,
                                                       float* __restrict__ rinv, int n) {
  const int row = blockIdx.x;
  const int t   = threadIdx.x;
  const float4* p = reinterpret_cast<const float4*>(C + (size_t)row * n);
  const int n4 = n >> 2;

  float m = -3.402823466e38f;
  for (int i = t; i < n4; i += 256) {
    float4 v = p[i];
    m = fmaxf(m, fmaxf(fmaxf(v.x, v.y), fmaxf(v.z, v.w)));
  }
#pragma unroll
  for (int off = 16; off > 0; off >>= 1) m = fmaxf(m, __shfl_xor(m, off, 32));
  __shared__ float redm[8];
  __shared__ float reds[8];
  if ((t & 31) == 0) redm[t >> 5] = m;
  __syncthreads();
  m = redm[0];
#pragma unroll
  for (int i = 1; i < 8; ++i) m = fmaxf(m, redm[i]);

  float s = 0.0f;
  for (int i = t; i < n4; i += 256) {
    float4 v = p[i];
    s += __expf(v.x - m) + __expf(v.y - m) + __expf(v.z - m) + __expf(v.w - m);
  }
#pragma unroll
  for (int off = 16; off > 0; off >>= 1) s += __shfl_xor(s, off, 32);
  if ((t & 31) == 0) reds[t >> 5] = s;
  __syncthreads();
  s = 0.0f;
#pragma unroll
  for (int i = 0; i < 8; ++i) s += reds[i];
  if (t == 0) { rmax[row] = m; rinv[row] = 1.0f / s; }
}

// ---------------------------------------------------------------------------
// 4) Fused attention GEMM: out[4096x128] = softmax(C) @ F.
//    Double-buffered K blocks of 32:
//      - F chunk staged via GLOBAL_LOAD_ASYNC_TO_LDS_B128 (ASYNCcnt),
//        awaited with s_wait_asynccnt AFTER the WMMA loop (overlap).
//      - exp(C - rowmax) chunk: global->regs before compute, exp+LDS store
//        after compute (latency hidden under WMMA).
//    LDS strides picked for conflict-free fragment reads:
//      Fl stride 136 (hi-half lanes shift banks by +16), Al stride 36.
// ---------------------------------------------------------------------------
#define KB 32
__global__ __launch_bounds__(256) void attn_gemm_wmma_kernel(const float* __restrict__ C,
                                                             const float* __restrict__ F,
                                                             const float* __restrict__ rmax,
                                                             const float* __restrict__ rinv,
                                                             float* __restrict__ out) {
  __shared__ __align__(16) float Al[2][64][36];    // exp tile [m][k]
  __shared__ __align__(16) float Fl[2][KB][136];   // F chunk  [k][n]

  const int tid  = threadIdx.x;
  const int w    = tid >> 5;
  const int lane = tid & 31;
  const int lo   = lane & 15;
  const int hi   = lane >> 4;
  const int wm   = (w & 3) * 16;
  const int wn   = (w >> 2) * 64;
  const int m0   = blockIdx.x * 64;

  // per-thread staging coordinates
  const int fk  = (tid + 0 * 256) >> 5;        // F: 4 reps, k = idx>>5, n4 = idx&31
  const int ar  = tid >> 3;                    // A: 2 reps, row = idx>>3, c4 = idx&7
  const int ac4 = tid & 7;

  v8f zero;
#pragma unroll
  for (int e = 0; e < 8; ++e) zero[e] = 0.0f;
  v8f acc[4];
#pragma unroll
  for (int ni = 0; ni < 4; ++ni) acc[ni] = zero;

  // ---- prologue: stage block 0 into buffer 0 ----
  {
#pragma unroll
    for (int j = 0; j < 4; ++j) {
      const int idx = tid + j * 256;
      const int k   = idx >> 5;
      const int n   = (idx & 31) * 4;
      const unsigned l = lds_off(&Fl[0][k][n]);
      const float* g  = F + (size_t)k * DDIM + n;
      asm volatile("global_load_async_to_lds_b128 %0, %1, off"
                   :: "v"(l), "v"(g) : "memory");
    }
#pragma unroll
    for (int j = 0; j < 2; ++j) {
      const int idx = tid + j * 256;
      const int row = idx >> 3;
      const int c4  = (idx & 7) * 4;
      float4 v = *reinterpret_cast<const float4*>(C + (size_t)(m0 + row) * NPOS + c4);
      const float mx = rmax[m0 + row];
      float4 e;
      e.x = __expf(v.x - mx); e.y = __expf(v.y - mx);
      e.z = __expf(v.z - mx); e.w = __expf(v.w - mx);
      *reinterpret_cast<float4*>(&Al[0][row][c4]) = e;
    }
    asm volatile("s_wait_asynccnt 0x0" ::: "memory");
    __syncthreads();
  }

  for (int kb = 0; kb < NPOS; kb += KB) {
    const int cur = (kb / KB) & 1;
    const int nxt = cur ^ 1;
    const bool has_next = (kb + KB) < NPOS;

    float4 av[2];
    if (has_next) {
      // issue async F stage for next block (lands while we do WMMA below)
#pragma unroll
      for (int j = 0; j < 4; ++j) {
        const int idx = tid + j * 256;
        const int k   = idx >> 5;
        const int n   = (idx & 31) * 4;
        const unsigned l = lds_off(&Fl[nxt][k][n]);
        const float* g  = F + (size_t)(kb + KB + k) * DDIM + n;
        asm volatile("global_load_async_to_lds_b128 %0, %1, off"
                     :: "v"(l), "v"(g) : "memory");
      }
      // issue A loads for next block into registers (no wait yet)
#pragma unroll
      for (int j = 0; j < 2; ++j) {
        const int idx = tid + j * 256;
        const int row = idx >> 3;
        const int c4  = (idx & 7) * 4;
        av[j] = *reinterpret_cast<const float4*>(
            C + (size_t)(m0 + row) * NPOS + kb + KB + c4);
      }
    }

    // ---- WMMA over current buffer ----
#pragma unroll
    for (int kc = 0; kc < KB; kc += 4) {
      const int kk = kc + 2 * hi;
      v2f a = *reinterpret_cast<const v2f*>(&Al[cur][wm + lo][kk]);
#pragma unroll
      for (int ni = 0; ni < 4; ++ni) {
        const int col = wn + 16 * ni + lo;
        v2f b;
        b.x = Fl[cur][kk][col];
        b.y = Fl[cur][kk + 1][col];
        acc[ni] = __builtin_amdgcn_wmma_f32_16x16x4_f32(
            false, a, false, b, (short)0, acc[ni], false, false);
      }
    }

    if (has_next) {
      // exp + LDS store of next A block (loads have landed by now)
#pragma unroll
      for (int j = 0; j < 2; ++j) {
        const int idx = tid + j * 256;
        const int row = idx >> 3;
        const int c4  = (idx & 7) * 4;
        const float mx = rmax[m0 + row];
        float4 e;
        e.x = __expf(av[j].x - mx); e.y = __expf(av[j].y - mx);
        e.z = __expf(av[j].z - mx); e.w = __expf(av[j].w - mx);
        *reinterpret_cast<float4*>(&Al[nxt][row][c4]) = e;
      }
      asm volatile("s_wait_asynccnt 0x0" ::: "memory");
    }
    __syncthreads();
  }

#pragma unroll
  for (int ni = 0; ni < 4; ++ni) {
    const int col = wn + 16 * ni + lo;
#pragma unroll
    for (int r = 0; r < 8; ++r) {
      const int row = m0 + wm + r + 8 * hi;
      out[(size_t)row * DDIM + col] = acc[ni][r] * rinv[row];
    }
  }
}

// ---------------------------------------------------------------------------
// 5) Direct 3x3 NHWC conv + bias + ReLU (SAME padding).
// ---------------------------------------------------------------------------
__global__ __launch_bounds__(256) void conv3x3_kernel(const float* __restrict__ in,
                                                      const float* __restrict__ wgt,
                                                      const float* __restrict__ bias,
                                                      float* __restrict__ out,
                                                      int H, int W, int Cin, int Cout,
                                                      int stride, int padBeg, int Ho, int Wo) {
  const int tid = blockIdx.x * blockDim.x + threadIdx.x;
  const int total = Ho * Wo * Cout;
  if (tid >= total) return;
  const int co = tid % Cout;
  const int xo = (tid / Cout) % Wo;
  const int yo = tid / (Cout * Wo);

  float acc = bias[co];
#pragma unroll
  for (int ky = 0; ky < 3; ++ky) {
    const int iy = yo * stride + ky - padBeg;
    if (iy < 0 || iy >= H) continue;
#pragma unroll
    for (int kx = 0; kx < 3; ++kx) {
      const int ix = xo * stride + kx - padBeg;
      if (ix < 0 || ix >= W) continue;
      const float* ip = in + ((size_t)iy * W + ix) * Cin;
      const float* wp = wgt + ((size_t)(ky * 3 + kx) * Cin) * Cout + co;
      int ci = 0;
      for (; ci + 4 <= Cin; ci += 4) {
        acc += ip[ci + 0] * wp[(size_t)(ci + 0) * Cout];
        acc += ip[ci + 1] * wp[(size_t)(ci + 1) * Cout];
        acc += ip[ci + 2] * wp[(size_t)(ci + 2) * Cout];
        acc += ip[ci + 3] * wp[(size_t)(ci + 3) * Cout];
      }
      for (; ci < Cin; ++ci) acc += ip[ci] * wp[(size_t)ci * Cout];
    }
  }
  out[tid] = fmaxf(acc, 0.0f);
}

// ---------------------------------------------------------------------------
// 6) Classification head
// ---------------------------------------------------------------------------
__global__ void colmax_partial_kernel(const float* __restrict__ x, float* __restrict__ pm,
                                      int rowsPer) {
  const int d = threadIdx.x;        // 128 threads
  const int b = blockIdx.x;         // 16 blocks
  float m = -3.402823466e38f;
  for (int r = b * rowsPer; r < (b + 1) * rowsPer; ++r)
    m = fmaxf(m, x[(size_t)r * DDIM + d]);
  pm[b * DDIM + d] = m;
}

__global__ void dense1_kernel(const float* __restrict__ pm, const float* __restrict__ w1,
                              const float* __restrict__ b1, float* __restrict__ g) {
  __shared__ float mp[DDIM];
  const int t = threadIdx.x;        // 512 threads
  if (t < DDIM) {
    float m = -3.402823466e38f;
    for (int b = 0; b < 16; ++b) m = fmaxf(m, pm[b * DDIM + t]);
    mp[t] = m;
  }
  __syncthreads();
  float acc = b1[t];
  for (int d = 0; d < DDIM; ++d) acc += mp[d] * w1[(size_t)d * 512 + t];
  g[t] = fmaxf(acc, 0.0f);
}

__global__ void dense2_softmax_kernel(const float* __restrict__ g, const float* __restrict__ w2,
                                      const float* __restrict__ b2, float* __restrict__ out) {
  __shared__ float gs[512];
  __shared__ float red[1024];
  const int t = threadIdx.x;        // 1024 threads
  if (t < 512) gs[t] = g[t];
  __syncthreads();
  float l = -3.402823466e38f;
  if (t < NCLS) {
    l = b2[t];
    for (int j = 0; j < 512; ++j) l += gs[j] * w2[(size_t)j * NCLS + t];
  }
  red[t] = l;
  __syncthreads();
  for (int s = 512; s > 0; s >>= 1) {
    if (t < s) red[t] = fmaxf(red[t], red[t + s]);
    __syncthreads();
  }
  const float mx = red[0];
  __syncthreads();
  const float e = (t < NCLS) ? __expf(l - mx) : 0.0f;
  red[t] = e;
  __syncthreads();
  for (int s = 512; s > 0; s >>= 1) {
    if (t < s) red[t] += red[t + s];
    __syncthreads();
  }
  const float inv = 1.0f / red[0];
  if (t < NCLS) out[t] = e * inv;
}

// ---------------------------------------------------------------------------
// launcher
// ---------------------------------------------------------------------------
extern "C" void kernel_launch(void* const* d_in, const int* in_sizes, int n_in,
                              void* d_out, int out_size, void* d_ws, size_t ws_size,
                              hipStream_t stream) {
  const float* t_lum  = (const float*)d_in[0];
  const float* r_lum  = (const float*)d_in[1];
  const float* r_ab   = (const float*)d_in[2];
  const float* w_rab1 = (const float*)d_in[3];
  const float* b_rab1 = (const float*)d_in[4];
  const float* w_rab2 = (const float*)d_in[5];
  const float* b_rab2 = (const float*)d_in[6];
  const float* w_fa1  = (const float*)d_in[7];
  const float* b_fa1  = (const float*)d_in[8];
  const float* w_fa2  = (const float*)d_in[9];
  const float* b_fa2  = (const float*)d_in[10];
  const float* w_fa3  = (const float*)d_in[11];
  const float* b_fa3  = (const float*)d_in[12];
  const float* w_d1   = (const float*)d_in[13];
  const float* b_d1   = (const float*)d_in[14];
  const float* w_d2   = (const float*)d_in[15];
  const float* b_d2   = (const float*)d_in[16];

  float* out = (float*)d_out;
  float* ws  = (float*)d_ws;

  float* tn   = ws + W_TN;
  float* rn   = ws + W_RN;
  float* f1   = ws + W_F1;
  float* F    = ws + W_F;
  float* fa   = ws + W_FA;
  float* rmax = ws + W_RMAX;
  float* rinv = ws + W_RINV;
  float* pm   = ws + W_PM;
  float* g    = ws + W_G;

  float* Cm   = out + O_C;

  // 1) normalize rows of t_lum / r_lum (8 rows per 256-thread block)
  rownorm_kernel<<<NPOS / 8, 256, 0, stream>>>(t_lum, tn, NPOS);
  rownorm_kernel<<<NPOS / 8, 256, 0, stream>>>(r_lum, rn, NPOS);

  // 2) C = tn @ rn^T  (fp32 WMMA), direct into d_out
  gemm_nt_wmma_kernel<<<dim3(NPOS / 128, NPOS / 128), 256, 0, stream>>>(tn, rn, Cm, NPOS);

  // 3) row softmax stats over C
  rowstats_kernel<<<NPOS, 256, 0, stream>>>(Cm, rmax, rinv, NPOS);

  // 4) reference-ab conv stack
  conv3x3_kernel<<<(128 * 128 * 64) / 256, 256, 0, stream>>>(
      r_ab, w_rab1, b_rab1, f1, 128, 128, 2, 64, 1, 1, 128, 128);
  conv3x3_kernel<<<(64 * 64 * 128) / 256, 256, 0, stream>>>(
      f1, w_rab2, b_rab2, F, 128, 128, 64, 128, 2, 0, 64, 64);

  // 5) fused attention GEMM: fa = softmax(C) @ F (WMMA + async-to-LDS)
  attn_gemm_wmma_kernel<<<NPOS / 64, 256, 0, stream>>>(Cm, F, rmax, rinv, fa);

  // 6) feature conv stack -> f_s3, f_s2, f_s1 directly into d_out
  conv3x3_kernel<<<(32 * 32 * 128) / 256, 256, 0, stream>>>(
      fa, w_fa1, b_fa1, out + O_FS3, 64, 64, 128, 128, 2, 0, 32, 32);
  conv3x3_kernel<<<(16 * 16 * 256) / 256, 256, 0, stream>>>(
      out + O_FS3, w_fa2, b_fa2, out + O_FS2, 32, 32, 128, 256, 2, 0, 16, 16);
  conv3x3_kernel<<<(8 * 8 * 512) / 256, 256, 0, stream>>>(
      out + O_FS2, w_fa3, b_fa3, out + O_FS1, 16, 16, 256, 512, 2, 0, 8, 8);

  // 7) head: column max-pool -> dense+relu -> dense+softmax
  colmax_partial_kernel<<<16, 128, 0, stream>>>(t_lum, pm, NPOS / 16);
  dense1_kernel<<<1, 512, 0, stream>>>(pm, w_d1, b_d1, g);
  dense2_softmax_kernel<<<1, 1024, 0, stream>>>(g, w_d2, b_d2, out + O_GTL);
}